// HeteroRGCN_86955907875557
// MI455X (gfx1250) — compile-verified
//
#include <hip/hip_runtime.h>
#include <hip/hip_bf16.h>

typedef __attribute__((ext_vector_type(2))) float v2f;
typedef __attribute__((ext_vector_type(8))) float v8f;

#define NNODES 100000
#define NEDGES 1000000
#define FDIM   64

// ---------------------------------------------------------------------------
// Zero a float region.
// ---------------------------------------------------------------------------
__global__ void zero_kernel(float* __restrict__ p, long long n) {
    long long i = (long long)blockIdx.x * blockDim.x + threadIdx.x;
    if (i < n) p[i] = 0.0f;
}

// ---------------------------------------------------------------------------
// Edge scatter: S[dst] += ew * x[src]   (and deg[dst] += ew if deg != null)
// 32 lanes per edge, 2 features per lane (b64 gather, two f32 atomics).
// x tables (25.6MB) and S (25.6MB) are L2-resident on MI455X (192MB L2).
// ---------------------------------------------------------------------------
__global__ void scatter_kernel(const float* __restrict__ xsrc,
                               const int*   __restrict__ src,
                               const int*   __restrict__ dst,
                               const float* __restrict__ ew,
                               float*       __restrict__ Sagg,
                               float*       __restrict__ deg,
                               int nEdges) {
    long long tid = (long long)blockIdx.x * blockDim.x + threadIdx.x;
    int e = (int)(tid >> 5);
    if (e >= nEdges) return;
    int lane = (int)(tid & 31);

    int s = src[e];
    int d = dst[e];
    float w = ew[e];

    const float2 xv = *(const float2*)(xsrc + (long long)s * FDIM + lane * 2);
    float* outp = Sagg + (long long)d * FDIM + lane * 2;
    atomicAdd(outp + 0, w * xv.x);
    atomicAdd(outp + 1, w * xv.y);
    if (deg != nullptr && lane == 0) atomicAdd(&deg[d], w);
}

// ---------------------------------------------------------------------------
// Dense epilogue GEMM with f32 WMMA (exact precision):
//   out = (Sagg @ Wlin) / max(deg,1) + Xdst @ Wroot + bias   [+ ReLU]
// One wave per 16x16 output tile; 4 waves/block cover the 64 output cols
// of one 16-row block. K=64 -> 16 x V_WMMA_F32_16X16X4_F32 per GEMM.
//
// A-fragment (16x4 f32, ISA 7.12.2): lane<16 holds (M=lane, K=4k..4k+1),
// lane>=16 holds (M=lane-16, K=4k+2..4k+3)  -> one aligned float2 load.
// B-fragment (4x16 f32): lane = column N, VGPR j holds row K=4k+2*(lane>>4)+j.
// C/D (16x16 f32): VGPR i, lane l -> row = i + 8*(l>>4), col = l&15.
// ---------------------------------------------------------------------------
__global__ void __launch_bounds__(128)
dense_wmma_kernel(const float* __restrict__ Sagg,   // [n,64] scatter accumulator
                  const float* __restrict__ deg,    // [n]
                  const float* __restrict__ Xdst,   // [n,64] root input
                  const float* __restrict__ Wlin,   // [64,64] row-major
                  const float* __restrict__ Wroot,  // [64,64] row-major
                  const float* __restrict__ bias,   // [64]
                  float*       __restrict__ outp,   // [n,64]
                  int relu) {
    const int lane = threadIdx.x & 31;
    const int wv   = threadIdx.x >> 5;         // column tile 0..3
    const int row0 = blockIdx.x * 16;
    const int col0 = wv * 16;
    const int half = lane >> 4;                // 0 or 1
    const int r    = lane & 15;                // row (A) or column (B/C)

    v8f accS = {};
    v8f accX = {};

    const float* aSrow = Sagg + (long long)(row0 + r) * FDIM;
    const float* aXrow = Xdst + (long long)(row0 + r) * FDIM;

#pragma unroll
    for (int k = 0; k < FDIM; k += 4) {
        const int kb = k + 2 * half;
        v2f aS = *(const v2f*)(aSrow + kb);
        v2f aX = *(const v2f*)(aXrow + kb);
        v2f bL = { Wlin [(kb + 0) * FDIM + col0 + r],
                   Wlin [(kb + 1) * FDIM + col0 + r] };
        v2f bR = { Wroot[(kb + 0) * FDIM + col0 + r],
                   Wroot[(kb + 1) * FDIM + col0 + r] };
        accS = __builtin_amdgcn_wmma_f32_16x16x4_f32(
            false, aS, false, bL, (short)0, accS, false, false);
        accX = __builtin_amdgcn_wmma_f32_16x16x4_f32(
            false, aX, false, bR, (short)0, accX, false, false);
    }

    const float b = bias[col0 + r];
#pragma unroll
    for (int i = 0; i < 8; ++i) {
        const int row = row0 + i + 8 * half;
        float d = deg[row];
        d = d > 1.0f ? d : 1.0f;
        float v = accS[i] / d + accX[i] + b;
        if (relu) v = v > 0.0f ? v : 0.0f;
        outp[(long long)row * FDIM + col0 + r] = v;
    }
}

// ---------------------------------------------------------------------------
// Launch: layer1 (scatter ab/ba -> dense+ReLU), layer2 (scatter -> dense).
// deg is identical in both layers (same edges/weights): computed once.
// ---------------------------------------------------------------------------
extern "C" void kernel_launch(void* const* d_in, const int* in_sizes, int n_in,
                              void* d_out, int out_size, void* d_ws, size_t ws_size,
                              hipStream_t stream) {
    const int N = NNODES, E = NEDGES, F = FDIM;

    const float* x_a        = (const float*)d_in[0];
    const float* x_b        = (const float*)d_in[1];
    const int*   ei_ab      = (const int*)  d_in[2];   // [2,E]: row0=src, row1=dst
    const int*   ei_ba      = (const int*)  d_in[3];
    const float* ew_ab      = (const float*)d_in[4];
    const float* ew_ba      = (const float*)d_in[5];
    const float* lin1_ab    = (const float*)d_in[6];
    const float* root1_ab_w = (const float*)d_in[7];
    const float* lin1_ba    = (const float*)d_in[8];
    const float* root1_ba_w = (const float*)d_in[9];
    const float* lin2_ab    = (const float*)d_in[10];
    const float* root2_ab_w = (const float*)d_in[11];
    const float* lin2_ba    = (const float*)d_in[12];
    const float* root2_ba_w = (const float*)d_in[13];
    const float* root1_ab_b = (const float*)d_in[14];
    const float* root1_ba_b = (const float*)d_in[15];
    const float* root2_ab_b = (const float*)d_in[16];
    const float* root2_ba_b = (const float*)d_in[17];

    // Workspace layout: [S_b | S_a | deg_b | deg_a | h_a | h_b]
    float* S_b   = (float*)d_ws;
    float* S_a   = S_b + (size_t)N * F;
    float* deg_b = S_a + (size_t)N * F;
    float* deg_a = deg_b + N;
    float* h_a   = deg_a + N;
    float* h_b   = h_a + (size_t)N * F;

    float* out_a = (float*)d_out;              // tuple order: (out_a, out_b)
    float* out_b = out_a + (size_t)N * F;

    const int ZB = 256;
    const dim3 sgrid((unsigned)(((long long)E * 32 + 255) / 256));
    const dim3 dgrid(N / 16);

    // --- zero S_b, S_a, deg_b, deg_a (contiguous region) ---
    long long nz = 2LL * N * F + 2LL * N;
    zero_kernel<<<(unsigned)((nz + ZB - 1) / ZB), ZB, 0, stream>>>(S_b, nz);

    // --- layer 1 ---
    scatter_kernel<<<sgrid, 256, 0, stream>>>(x_a, ei_ab, ei_ab + E, ew_ab, S_b, deg_b, E);
    scatter_kernel<<<sgrid, 256, 0, stream>>>(x_b, ei_ba, ei_ba + E, ew_ba, S_a, deg_a, E);
    dense_wmma_kernel<<<dgrid, 128, 0, stream>>>(S_b, deg_b, x_b, lin1_ab, root1_ab_w,
                                                 root1_ab_b, h_b, 1);
    dense_wmma_kernel<<<dgrid, 128, 0, stream>>>(S_a, deg_a, x_a, lin1_ba, root1_ba_w,
                                                 root1_ba_b, h_a, 1);

    // --- re-zero S_b, S_a for layer 2 ---
    long long nz2 = 2LL * N * F;
    zero_kernel<<<(unsigned)((nz2 + ZB - 1) / ZB), ZB, 0, stream>>>(S_b, nz2);

    // --- layer 2 (deg unchanged) ---
    scatter_kernel<<<sgrid, 256, 0, stream>>>(h_a, ei_ab, ei_ab + E, ew_ab, S_b, nullptr, E);
    scatter_kernel<<<sgrid, 256, 0, stream>>>(h_b, ei_ba, ei_ba + E, ew_ba, S_a, nullptr, E);
    dense_wmma_kernel<<<dgrid, 128, 0, stream>>>(S_b, deg_b, h_b, lin2_ab, root2_ab_w,
                                                 root2_ab_b, out_b, 0);
    dense_wmma_kernel<<<dgrid, 128, 0, stream>>>(S_a, deg_a, h_a, lin2_ba, root2_ba_w,
                                                 root2_ba_b, out_a, 0);
}